// Mypenalizer_13022340841535
// MI455X (gfx1250) — compile-verified
//
#include <hip/hip_runtime.h>
#include <cstdint>

// Penalizer: out[i] = (0 <= X[i,7] <= 1) ? 0.01*exp(4.81*(1+X[i,7])) : 3.0
// Memory-bound: every 128B line of X is touched (tails at 32B stride), so
// ~576MB of HBM traffic -> ~25us floor at 23.3 TB/s. Kernel = pure streaming.

#define TPB   256
#define ITERS 8   // rows per thread; block covers TPB*ITERS = 2048 rows

#define A_COEF   0.01f
#define B_COEF   4.81f
#define OOB_COST 3.0f

#if defined(__HIP_DEVICE_COMPILE__) &&                                   \
    __has_builtin(__builtin_amdgcn_global_load_async_to_lds_b32) &&      \
    __has_builtin(__builtin_amdgcn_s_wait_asynccnt)
#define ASYNC_OK 1
#else
#define ASYNC_OK 0
#if defined(__HIP_DEVICE_COMPILE__)
#warning "gfx1250 async-to-LDS builtins unavailable on this toolchain; using direct nontemporal-load fallback"
#endif
#endif

// Builtin signature (per hipcc diagnostic): (global int*, local int*, imm, imm)
typedef __attribute__((address_space(1))) int g_i32_t;
typedef __attribute__((address_space(3))) int l_i32_t;

__device__ __forceinline__ float penal(float v) {
  // fast hw exp path (v_exp_f32); accuracy ~few ulp, fine for this cost fn
  float cost = A_COEF * __expf(B_COEF * (1.0f + v));
  return (v >= 0.0f && v <= 1.0f) ? cost : OOB_COST;
}

__global__ void __launch_bounds__(TPB)
penal_kernel(const float* __restrict__ X, float* __restrict__ out, int n) {
#if ASYNC_OK
  // Double-buffered async gather of row tails into LDS. Each thread owns one
  // 4B slot per buffer; async loads complete in order, so s_wait_asynccnt is
  // the only sync needed (no cross-thread LDS sharing -> no barrier).
  __shared__ float buf[2][TPB];
  const int tid  = (int)threadIdx.x;
  const int base = (int)blockIdx.x * (TPB * ITERS) + tid;

  l_i32_t* slot0 = (l_i32_t*)(unsigned int)(uintptr_t)&buf[0][tid];
  l_i32_t* slot1 = (l_i32_t*)(unsigned int)(uintptr_t)&buf[1][tid];

  // prologue: stage iteration 0 into buffer 0
  {
    int r = base; if (r >= n) r = n - 1;
    const float* gp = X + (size_t)r * 8 + 7;
    __builtin_amdgcn_global_load_async_to_lds_b32(
        (g_i32_t*)(uintptr_t)gp, slot0, /*offset=*/0, /*cpol=*/0);
  }

#pragma unroll
  for (int it = 0; it < ITERS; ++it) {
    if (it + 1 < ITERS) {
      // issue next stage into the other buffer, then wait for current stage
      int rn = base + (it + 1) * TPB; if (rn >= n) rn = n - 1;
      const float* gp = X + (size_t)rn * 8 + 7;
      __builtin_amdgcn_global_load_async_to_lds_b32(
          (g_i32_t*)(uintptr_t)gp,
          ((it + 1) & 1) ? slot1 : slot0, /*offset=*/0, /*cpol=*/0);
      __builtin_amdgcn_s_wait_asynccnt(1);   // current buffer landed (in-order)
    } else {
      __builtin_amdgcn_s_wait_asynccnt(0);   // drain last stage
    }
    asm volatile("" ::: "memory");           // keep ds_load below the wait

    float v = buf[it & 1][tid];              // ds_load_b32 of own slot
    int r = base + it * TPB;
    if (r < n) __builtin_nontemporal_store(penal(v), out + r);
  }
#else
  // Fallback: direct strided nontemporal loads. Same grid geometry.
  const int base = (int)blockIdx.x * (TPB * ITERS) + (int)threadIdx.x;
#pragma unroll
  for (int it = 0; it < ITERS; ++it) {
    int r = base + it * TPB;
    if (r < n) {
      float v = __builtin_nontemporal_load(X + (size_t)r * 8 + 7);
      __builtin_nontemporal_store(penal(v), out + r);
    }
  }
#endif
}

extern "C" void kernel_launch(void* const* d_in, const int* in_sizes, int n_in,
                              void* d_out, int out_size, void* d_ws, size_t ws_size,
                              hipStream_t stream) {
  const float* X = (const float*)d_in[0];
  float* out    = (float*)d_out;
  const int n   = out_size;                 // 16,777,216 rows
  const int rows_per_block = TPB * ITERS;   // 2048
  const int blocks = (n + rows_per_block - 1) / rows_per_block;  // 8192
  penal_kernel<<<blocks, TPB, 0, stream>>>(X, out, n);
}